// GATLayer_67740224192705
// MI455X (gfx1250) — compile-verified
//
#include <hip/hip_runtime.h>

#define NN 50000
#define EE 1600000
#define DD 128
#define HH 4
#define SLOPE 0.2f

typedef __attribute__((ext_vector_type(2))) float v2f;
typedef __attribute__((ext_vector_type(8))) float v8f;

// Order-preserving float <-> uint mapping for atomicMax-based segment max.
__device__ __forceinline__ unsigned fmax_map(float f) {
  unsigned b = __float_as_uint(f);
  return b ^ (((unsigned)((int)b >> 31)) | 0x80000000u);
}
__device__ __forceinline__ float fmax_unmap(unsigned u) {
  unsigned b = (u & 0x80000000u) ? (u ^ 0x80000000u) : ~u;
  return __uint_as_float(b);
}
__device__ __forceinline__ float lrelu(float x) { return x >= 0.f ? x : SLOPE * x; }

// ---------------------------------------------------------------------------
// Kernel 1: per-node attention scalars. One wave (32 lanes) per node.
// s_self[n][h] = sum_d feat[n,d]*sk[h,d]*ak[h,d]
// s_nbr [n][h] = sum_d feat[n,d]*sk[h,d]*ak[h,D+d]
// ---------------------------------------------------------------------------
__global__ void __launch_bounds__(256)
gat_scores(const float* __restrict__ feat, const float* __restrict__ selfk,
           const float* __restrict__ attnk, float* __restrict__ sc_self,
           float* __restrict__ sc_nbr) {
  int node = (int)((blockIdx.x * blockDim.x + threadIdx.x) >> 5);
  int lane = threadIdx.x & 31;
  if (node >= NN) return;  // wave-uniform
  int d0 = lane * 4;
  float4 f = *(const float4*)(feat + (size_t)node * DD + d0);
  float ps[HH], pn[HH];
#pragma unroll
  for (int h = 0; h < HH; ++h) {
    float4 sk = *(const float4*)(selfk + h * DD + d0);
    float4 as_ = *(const float4*)(attnk + h * 2 * DD + d0);
    float4 an = *(const float4*)(attnk + h * 2 * DD + DD + d0);
    float hx = f.x * sk.x, hy = f.y * sk.y, hz = f.z * sk.z, hw = f.w * sk.w;
    ps[h] = hx * as_.x + hy * as_.y + hz * as_.z + hw * as_.w;
    pn[h] = hx * an.x + hy * an.y + hz * an.z + hw * an.w;
  }
#pragma unroll
  for (int off = 16; off >= 1; off >>= 1) {
#pragma unroll
    for (int h = 0; h < HH; ++h) {
      ps[h] += __shfl_xor(ps[h], off, 32);
      pn[h] += __shfl_xor(pn[h], off, 32);
    }
  }
  if (lane == 0) {
    *(float4*)(sc_self + (size_t)node * 4) = make_float4(ps[0], ps[1], ps[2], ps[3]);
    *(float4*)(sc_nbr + (size_t)node * 4) = make_float4(pn[0], pn[1], pn[2], pn[3]);
  }
}

// ---------------------------------------------------------------------------
// Kernel 2: segment max over edges (per src node, per head) via atomicMax.
// ---------------------------------------------------------------------------
__global__ void __launch_bounds__(256)
gat_edge_max(const long long* __restrict__ ei, const float* __restrict__ sc_self,
             const float* __restrict__ sc_nbr, unsigned* __restrict__ mmax) {
  int e = blockIdx.x * blockDim.x + threadIdx.x;
  if (e >= EE) return;
  int s = (int)ei[2 * (size_t)e];
  int t = (int)ei[2 * (size_t)e + 1];
  float4 a = *(const float4*)(sc_self + (size_t)s * 4);
  float4 b = *(const float4*)(sc_nbr + (size_t)t * 4);
  atomicMax(mmax + (size_t)s * 4 + 0, fmax_map(lrelu(a.x + b.x)));
  atomicMax(mmax + (size_t)s * 4 + 1, fmax_map(lrelu(a.y + b.y)));
  atomicMax(mmax + (size_t)s * 4 + 2, fmax_map(lrelu(a.z + b.z)));
  atomicMax(mmax + (size_t)s * 4 + 3, fmax_map(lrelu(a.w + b.w)));
}

// ---------------------------------------------------------------------------
// Kernel 3: denom[s][h] = sum_e exp(e_h - m[s][h]) via float atomicAdd.
// ---------------------------------------------------------------------------
__global__ void __launch_bounds__(256)
gat_edge_sum(const long long* __restrict__ ei, const float* __restrict__ sc_self,
             const float* __restrict__ sc_nbr, const unsigned* __restrict__ mmax,
             float* __restrict__ denom) {
  int e = blockIdx.x * blockDim.x + threadIdx.x;
  if (e >= EE) return;
  int s = (int)ei[2 * (size_t)e];
  int t = (int)ei[2 * (size_t)e + 1];
  float4 a = *(const float4*)(sc_self + (size_t)s * 4);
  float4 b = *(const float4*)(sc_nbr + (size_t)t * 4);
  uint4 mu = *(const uint4*)(mmax + (size_t)s * 4);
  atomicAdd(denom + (size_t)s * 4 + 0, expf(lrelu(a.x + b.x) - fmax_unmap(mu.x)));
  atomicAdd(denom + (size_t)s * 4 + 1, expf(lrelu(a.y + b.y) - fmax_unmap(mu.y)));
  atomicAdd(denom + (size_t)s * 4 + 2, expf(lrelu(a.z + b.z) - fmax_unmap(mu.z)));
  atomicAdd(denom + (size_t)s * 4 + 3, expf(lrelu(a.w + b.w) - fmax_unmap(mu.w)));
}

// ---------------------------------------------------------------------------
// Kernel 4: fused 4-head aggregation. One wave per edge.
// acc[s,d] += feat[t,d] * sum_h alpha_h * sk_h[d]   (mean over H deferred)
// ---------------------------------------------------------------------------
__global__ void __launch_bounds__(256)
gat_aggregate(const long long* __restrict__ ei, const float* __restrict__ feat,
              const float* __restrict__ selfk, const float* __restrict__ sc_self,
              const float* __restrict__ sc_nbr, const unsigned* __restrict__ mmax,
              const float* __restrict__ denom, float* __restrict__ acc) {
  int e = (int)((blockIdx.x * blockDim.x + threadIdx.x) >> 5);
  int lane = threadIdx.x & 31;
  if (e >= EE) return;  // wave-uniform
  int s = (int)ei[2 * (size_t)e];
  int t = (int)ei[2 * (size_t)e + 1];
  float4 a = *(const float4*)(sc_self + (size_t)s * 4);
  float4 b = *(const float4*)(sc_nbr + (size_t)t * 4);
  uint4 mu = *(const uint4*)(mmax + (size_t)s * 4);
  float4 dn = *(const float4*)(denom + (size_t)s * 4);
  float w0 = expf(lrelu(a.x + b.x) - fmax_unmap(mu.x)) / dn.x;
  float w1 = expf(lrelu(a.y + b.y) - fmax_unmap(mu.y)) / dn.y;
  float w2 = expf(lrelu(a.z + b.z) - fmax_unmap(mu.z)) / dn.z;
  float w3 = expf(lrelu(a.w + b.w) - fmax_unmap(mu.w)) / dn.w;

  int d0 = lane * 4;
  float4 f = *(const float4*)(feat + (size_t)t * DD + d0);
  float4 k0 = *(const float4*)(selfk + 0 * DD + d0);
  float4 k1 = *(const float4*)(selfk + 1 * DD + d0);
  float4 k2 = *(const float4*)(selfk + 2 * DD + d0);
  float4 k3 = *(const float4*)(selfk + 3 * DD + d0);
  float cx = w0 * k0.x + w1 * k1.x + w2 * k2.x + w3 * k3.x;
  float cy = w0 * k0.y + w1 * k1.y + w2 * k2.y + w3 * k3.y;
  float cz = w0 * k0.z + w1 * k1.z + w2 * k2.z + w3 * k3.z;
  float cw = w0 * k0.w + w1 * k1.w + w2 * k2.w + w3 * k3.w;
  float* o = acc + (size_t)s * DD + d0;
  atomicAdd(o + 0, f.x * cx);
  atomicAdd(o + 1, f.y * cy);
  atomicAdd(o + 2, f.z * cz);
  atomicAdd(o + 3, f.w * cw);
}

// ---------------------------------------------------------------------------
// Kernel 5: gate GEMM with V_WMMA_F32_16X16X4_F32 + fused finalize.
// One wave computes one 16x16 tile of Z = feat @ gate_weight, then applies
// bias+sigmoid and blends: out = relu(acc/H)*gate + feat*(1-gate).
// `io` holds the edge accumulator on entry and the final output on exit.
// ---------------------------------------------------------------------------
__global__ void __launch_bounds__(128)
gat_gate_finalize(const float* __restrict__ feat, const float* __restrict__ gw,
                  const float* __restrict__ gb, float* __restrict__ io) {
  int wid = threadIdx.x >> 5;
  int lane = threadIdx.x & 31;
  int tile = blockIdx.x * 4 + wid;      // 0 .. 24999
  int tileM = tile >> 3;                // 0 .. 3124
  int tileN = tile & 7;                 // 0 .. 7
  int half = lane >> 4;                 // 0 | 1 -> K offset 0 | 2
  int l15 = lane & 15;
  int m_a = tileM * 16 + l15;           // A-matrix row this lane supplies
  int n = tileN * 16 + l15;             // B/C column this lane supplies

  v8f c = {};
#pragma unroll
  for (int kk = 0; kk < DD / 4; ++kk) {
    int kb = kk * 4 + half * 2;
    // A (16x4): lane -> A[m_a, kb], A[m_a, kb+1]
    v2f av = *(const v2f*)(feat + (size_t)m_a * DD + kb);
    // B (4x16): lane -> B[kb, n], B[kb+1, n]
    v2f bv;
    bv[0] = gw[(size_t)kb * DD + n];
    bv[1] = gw[(size_t)(kb + 1) * DD + n];
    c = __builtin_amdgcn_wmma_f32_16x16x4_f32(
        /*neg_a=*/false, av, /*neg_b=*/false, bv,
        /*c_mod=*/(short)0, c, /*reuse_a=*/false, /*reuse_b=*/false);
  }

  float bias = gb[n];
#pragma unroll
  for (int i = 0; i < 8; ++i) {
    int m = tileM * 16 + i + half * 8;  // C/D: VGPR i -> rows i (lanes 0-15), i+8 (16-31)
    size_t idx = (size_t)m * DD + n;
    float z = c[i] + bias;
    float gate = 1.f / (1.f + expf(-z));
    float o = io[idx] * (1.f / (float)HH);  // mean over heads
    o = o > 0.f ? o : 0.f;                  // relu
    float fv = feat[idx];
    io[idx] = o * gate + fv * (1.f - gate);
  }
}

// ---------------------------------------------------------------------------
extern "C" void kernel_launch(void* const* d_in, const int* in_sizes, int n_in,
                              void* d_out, int out_size, void* d_ws, size_t ws_size,
                              hipStream_t stream) {
  const long long* ei = (const long long*)d_in[0];  // edge_index int64 (E,2)
  const float* feat = (const float*)d_in[1];        // (N,128)
  const float* selfk = (const float*)d_in[2];       // (4,128)
  const float* attnk = (const float*)d_in[3];       // (4,256)
  const float* gw = (const float*)d_in[4];          // (128,128)
  const float* gb = (const float*)d_in[5];          // (128,)
  float* out = (float*)d_out;                       // (N,128), also accumulator

  float* sc_self = (float*)d_ws;                          // N*4 floats
  float* sc_nbr = sc_self + (size_t)NN * 4;               // N*4 floats
  unsigned* mmax = (unsigned*)(sc_nbr + (size_t)NN * 4);  // N*4 uints
  float* denom = (float*)(mmax + (size_t)NN * 4);         // N*4 floats

  // Zero accumulator + (mmax=0 encodes -inf under the monotone map, denom=0).
  hipMemsetAsync(d_out, 0, (size_t)NN * DD * sizeof(float), stream);
  hipMemsetAsync(d_ws, 0, (size_t)NN * 16 * sizeof(float), stream);

  gat_scores<<<(NN + 7) / 8, 256, 0, stream>>>(feat, selfk, attnk, sc_self, sc_nbr);
  gat_edge_max<<<(EE + 255) / 256, 256, 0, stream>>>(ei, sc_self, sc_nbr, mmax);
  gat_edge_sum<<<(EE + 255) / 256, 256, 0, stream>>>(ei, sc_self, sc_nbr, mmax, denom);
  gat_aggregate<<<(EE + 7) / 8, 256, 0, stream>>>(ei, feat, selfk, sc_self, sc_nbr,
                                                  mmax, denom, out);
  gat_gate_finalize<<<(NN / 16) * (DD / 16) / 4, 128, 0, stream>>>(feat, gw, gb, out);
}